// Encoder_21526376087564
// MI455X (gfx1250) — compile-verified
//
#include <hip/hip_runtime.h>
#include <hip/hip_bf16.h>
#include <math.h>

// CDNA5 / gfx1250, wave32.
typedef float v2f __attribute__((ext_vector_type(2)));
typedef float v8f __attribute__((ext_vector_type(8)));

#define HID   2048
#define GATES 8192          // 4*HID
#define KSPAN 512           // K-slice per wave (HID / 4 waves)

// Accumulate c += W_tile(16 x KSPAN) @ v(KSPAN), replicated over N.
// wrow: per-lane pointer into weights (A layout: lane L<16 -> row L, k+0/1;
//       lane L+16 -> row L, k+2/3).  vk: per-lane pointer into LDS vector.
__device__ __forceinline__ void mm_accum(const float* __restrict__ wrow,
                                         const float* vk, v8f& c) {
    for (int k0 = 0; k0 < KSPAN; k0 += 32) {
        // Stream-prefetch 1KB ahead per row (stays inside this wave's slice).
        if (k0 + 288 <= KSPAN)
            __builtin_prefetch(wrow + k0 + 256, 0, 0);
#pragma unroll
        for (int kk = 0; kk < 32; kk += 4) {
            v2f a = *(const v2f*)(wrow + k0 + kk);   // global_load_b64
            v2f b = *(const v2f*)(vk + k0 + kk);     // ds_load_b64 broadcast
            c = __builtin_amdgcn_wmma_f32_16x16x4_f32(
                    /*neg_a=*/false, a, /*neg_b=*/false, b,
                    /*c_mod=*/(short)0, c, /*reuse_a=*/false, /*reuse_b=*/false);
        }
    }
}

// ---------------------------------------------------------------------------
// Kernel 1: gates[dir][0..8191] = W_ih[dir] @ x + W_hh[dir] @ h[dir]
// 1024 blocks x 128 threads. Block = one 16-row gate tile; its 4 waves each
// own a 512-float K-slice (of both W_ih and W_hh) -> 4096 independent waves
// for HBM latency hiding; partials reduced through LDS.
// ---------------------------------------------------------------------------
__global__ __launch_bounds__(128) void lstm_gates_wmma(
    const int* __restrict__ tok,
    const float* __restrict__ emb,
    const float* __restrict__ hid,        // (2,1,H)
    const float* __restrict__ Wih_f, const float* __restrict__ Whh_f,
    const float* __restrict__ Wih_b, const float* __restrict__ Whh_b,
    float* __restrict__ gates)            // (2, 8192)
{
    __shared__ float xbuf[HID];
    __shared__ float hbuf[HID];
    __shared__ float redbuf[4][16];

    const int tid  = threadIdx.x;        // 0..127
    const int wave = tid >> 5;           // 0..3  (K-slice owner)
    const int lane = tid & 31;
    const int dir  = blockIdx.x >> 9;    // 512 blocks per direction
    const int tile = blockIdx.x & 511;   // 16-row tile within direction

    // Stage x = emb[token] and h = hidden[dir] into LDS (shared by 4 waves).
    const int token = tok[0];            // low word valid for int32/int64 token
    const float* xsrc = emb + (size_t)token * HID;
    const float* hsrc = hid + (size_t)dir * HID;
    for (int i = tid; i < HID; i += 128) {
        xbuf[i] = xsrc[i];
        hbuf[i] = hsrc[i];
    }
    __syncthreads();

    const int row0 = tile * 16;          // base gate row (of 8192)
    const float* Wih = dir ? Wih_b : Wih_f;
    const float* Whh = dir ? Whh_b : Whh_f;

    const int mrow  = lane & 15;         // A-matrix row owned by this lane
    const int khalf = (lane >> 4) * 2;   // lanes 16-31 own K+2,K+3
    const int kbase = wave * KSPAN;      // this wave's K-slice

    const float* wrow_ih = Wih + (size_t)(row0 + mrow) * HID + kbase + khalf;
    const float* wrow_hh = Whh + (size_t)(row0 + mrow) * HID + kbase + khalf;
    const float* xk = xbuf + kbase + khalf;
    const float* hk = hbuf + kbase + khalf;

    v8f c = {0.f, 0.f, 0.f, 0.f, 0.f, 0.f, 0.f, 0.f};
    mm_accum(wrow_ih, xk, c);            // partial W_ih @ x
    mm_accum(wrow_hh, hk, c);            // partial W_hh @ h

    // Column n=0 of D lives in lane 0 (rows 0..7 in c[0..7]) and lane 16
    // (rows 8..15).  Dump 16 partials per wave, then reduce over the 4 slices.
    if ((lane & 15) == 0) {
        float* r = &redbuf[wave][(lane >> 4) * 8];
        r[0] = c[0]; r[1] = c[1]; r[2] = c[2]; r[3] = c[3];
        r[4] = c[4]; r[5] = c[5]; r[6] = c[6]; r[7] = c[7];
    }
    __syncthreads();
    if (tid < 16) {
        float v = redbuf[0][tid] + redbuf[1][tid] + redbuf[2][tid] + redbuf[3][tid];
        gates[(size_t)dir * GATES + row0 + tid] = v;
    }
}

// ---------------------------------------------------------------------------
// Kernel 2: biases + LSTM pointwise math + output assembly.
// d_out flat layout: [h_f,h_b | h_f,h_b | c_f,c_b]  (out, hidden_out, cell_out)
// ---------------------------------------------------------------------------
__global__ __launch_bounds__(256) void lstm_pointwise(
    const float* __restrict__ gates,
    const float* __restrict__ bih_f, const float* __restrict__ bhh_f,
    const float* __restrict__ bih_b, const float* __restrict__ bhh_b,
    const float* __restrict__ cell,   // (2,1,H)
    float* __restrict__ out)          // 12288 floats
{
    int idx = blockIdx.x * blockDim.x + threadIdx.x;   // 0..4095
    if (idx >= 2 * HID) return;
    int dir = idx >> 11;
    int j   = idx & (HID - 1);

    const float* g   = gates + (size_t)dir * GATES;
    const float* bih = dir ? bih_b : bih_f;
    const float* bhh = dir ? bhh_b : bhh_f;

    float gi = g[j]           + bih[j]           + bhh[j];
    float gf = g[HID + j]     + bih[HID + j]     + bhh[HID + j];
    float gg = g[2 * HID + j] + bih[2 * HID + j] + bhh[2 * HID + j];
    float go = g[3 * HID + j] + bih[3 * HID + j] + bhh[3 * HID + j];

    float i_ = 1.f / (1.f + expf(-gi));
    float f_ = 1.f / (1.f + expf(-gf));
    float g_ = tanhf(gg);
    float o_ = 1.f / (1.f + expf(-go));

    float c0 = cell[(size_t)dir * HID + j];
    float c2 = f_ * c0 + i_ * g_;
    float h2 = o_ * tanhf(c2);

    out[idx]        = h2;   // out        (1,1,2H)
    out[4096 + idx] = h2;   // hidden_out (2,1,H)
    out[8192 + idx] = c2;   // cell_out   (2,1,H)
}

extern "C" void kernel_launch(void* const* d_in, const int* in_sizes, int n_in,
                              void* d_out, int out_size, void* d_ws, size_t ws_size,
                              hipStream_t stream) {
    const int*   tok   = (const int*)d_in[0];
    const float* hid   = (const float*)d_in[1];
    const float* cell  = (const float*)d_in[2];
    const float* emb   = (const float*)d_in[3];
    const float* Wih_f = (const float*)d_in[4];
    const float* Whh_f = (const float*)d_in[5];
    const float* bih_f = (const float*)d_in[6];
    const float* bhh_f = (const float*)d_in[7];
    const float* Wih_b = (const float*)d_in[8];
    const float* Whh_b = (const float*)d_in[9];
    const float* bih_b = (const float*)d_in[10];
    const float* bhh_b = (const float*)d_in[11];

    float* gates = (float*)d_ws;          // 2 * 8192 floats = 64 KB scratch
    float* out   = (float*)d_out;         // 12288 floats

    lstm_gates_wmma<<<1024, 128, 0, stream>>>(tok, emb, hid,
                                              Wih_f, Whh_f, Wih_b, Whh_b, gates);
    lstm_pointwise<<<16, 256, 0, stream>>>(gates, bih_f, bhh_f, bih_b, bhh_b,
                                           cell, out);
}